// DVQBottleneck_34359738654
// MI455X (gfx1250) — compile-verified
//
#include <hip/hip_runtime.h>

typedef float v2f __attribute__((ext_vector_type(2)));
typedef float v4f __attribute__((ext_vector_type(4)));
typedef float v8f __attribute__((ext_vector_type(8)));

#define B_    8
#define N_    4096
#define D_    1024
#define S_    4
#define K_    16
#define d_    256
#define TOKENS (B_ * N_)           // 32768
#define BETA  0.25f
#define NBLOCKS (TOKENS / 32)      // 1024 (2 tiles of 16 tokens per block)
#define NPART (NBLOCKS * 8)        // 8192 per-wave loss partials

// ---------------------------------------------------------------------------
// Pass 0: codebook squared norms  cnorm[s*16+k] = sum_j codebooks[s][k][j]^2
// ---------------------------------------------------------------------------
__global__ void dvq_cnorm_kernel(const float* __restrict__ cb,
                                 float* __restrict__ cnorm) {
    int i = threadIdx.x;                    // 64 threads: one per (s,k)
    const float* row = cb + (size_t)i * d_;
    float acc = 0.f;
    for (int j = 0; j < d_; ++j) acc += row[j] * row[j];
    cnorm[i] = acc;
}

// ---------------------------------------------------------------------------
// Pass 1: WMMA distance matrix + argmin + loss partial + z/ids outputs.
// Block = 256 threads = 8 waves. Wave (t,s): token tile t (of 2), subvec s.
// ---------------------------------------------------------------------------
__global__ __launch_bounds__(256) void dvq_main_kernel(
    const float* __restrict__ h, const float* __restrict__ cb,
    const float* __restrict__ cnorm, float* __restrict__ z_out,
    int* __restrict__ ids_out, float* __restrict__ partials) {
    __shared__ int ids_lds[2][16][4];       // [tile][row][s]

    const int tid  = threadIdx.x;
    const int wave = tid >> 5;
    const int lane = tid & 31;
    const int t    = wave >> 2;             // 0..1 token tile in block
    const int s    = wave & 3;              // 0..3 subvector
    const int m    = lane & 15;             // row (A) / codeword (B) index
    const int hi   = lane >> 4;             // lane-half selects K pair

    const int tile_base = (blockIdx.x * 2 + t) * 16;
    const int row = tile_base + m;

    // A: 16x4 f32 tile of ze; lane l, vgpr v holds A[l&15][(l>>4)*2 + v]
    const float* aptr = h  + (size_t)row * D_ + s * d_ + hi * 2;
    // B: 4x16 f32 codebook slice; lane l, vgpr v holds B[(l>>4)*2+v][l&15]
    const float* bptr = cb + ((size_t)s * K_ + m) * d_ + hi * 2;

    v8f acc = {};
    float anorm = 0.f;
#pragma unroll 4
    for (int kc = 0; kc < d_ / 4; ++kc) {
        v2f a = *(const v2f*)(aptr + kc * 4);
        v2f b = *(const v2f*)(bptr + kc * 4);
        anorm += a.x * a.x + a.y * a.y;
        // D = A(16x4) * B(4x16) + C ; emits v_wmma_f32_16x16x4_f32
        acc = __builtin_amdgcn_wmma_f32_16x16x4_f32(
            false, a, false, b, (short)0, acc, false, false);
    }
    // Full row norm: lo lanes held cols {0,1 mod4}, hi lanes {2,3 mod4}
    float zenorm = anorm + __shfl_xor(anorm, 16, 32);   // row (lane&15) norm

    // score[m][k] = ||c_k||^2 - 2 ze.c_k  (ties -> smaller k, like argmin)
    const float cn = cnorm[s * 16 + m];
    float bestv[8];
    int   besti[8];
#pragma unroll
    for (int r = 0; r < 8; ++r) { bestv[r] = cn - 2.0f * acc[r]; besti[r] = m; }
#pragma unroll
    for (int mask = 1; mask <= 8; mask <<= 1) {
#pragma unroll
        for (int r = 0; r < 8; ++r) {
            float ov = __shfl_xor(bestv[r], mask, 32);
            int   oi = __shfl_xor(besti[r], mask, 32);
            if (ov < bestv[r] || (ov == bestv[r] && oi < besti[r])) {
                bestv[r] = ov; besti[r] = oi;
            }
        }
    }

    // Loss partial: sum_m (best_score[m] + ||ze_m||^2)  ==  sum_m ||zq-ze||^2
    float w = 0.f;
#pragma unroll
    for (int r = 0; r < 8; ++r) w += bestv[r];
    float cross = w + __shfl_xor(w, 16, 32);            // all 16 rows' best
    float q = zenorm;
#pragma unroll
    for (int mask = 1; mask <= 8; mask <<= 1) q += __shfl_xor(q, mask, 32);
    if (lane == 0) partials[blockIdx.x * 8 + wave] = q + cross;

    // Publish ids: lane 0 covers rows 0..7, lane 16 rows 8..15
    if (m == 0) {
#pragma unroll
        for (int r = 0; r < 8; ++r) ids_lds[t][r + 8 * hi][s] = besti[r];
    }
    __syncthreads();

    // ids_packed
    if (tid < 32) {
        int tt = tid >> 4, mm = tid & 15;
        int token = (blockIdx.x * 2 + tt) * 16 + mm;
        int packed = 0;
#pragma unroll
        for (int ss = 0; ss < 4; ++ss) packed |= ids_lds[tt][mm][ss] << (4 * ss);
        ids_out[token] = packed;
    }

    // z = zq : vectorized (b128) coalesced gather of codebook rows.
    // 32 tokens x 1024 cols = 8192 float4; 256 threads -> 32 iterations.
    const v4f* __restrict__ cb4 = (const v4f*)cb;
    v4f* __restrict__ z4 = (v4f*)(z_out + (size_t)blockIdx.x * 32 * D_);
#pragma unroll 4
    for (int it = 0; it < 32; ++it) {
        int e4   = tid + it * 256;          // 0..8191 float4 index
        int tl   = e4 >> 8;                 // local token 0..31 (256 f4/row)
        int col4 = e4 & 255;
        int ss   = col4 >> 6, j4 = col4 & 63;   // 64 float4 per subvector
        int id   = ids_lds[tl >> 4][tl & 15][ss];
        z4[e4] = cb4[(size_t)(ss * 16 + id) * 64 + j4];
    }
}

// ---------------------------------------------------------------------------
// Pass 2: deterministic fixed-order reduction of 8192 partials -> scalar.
// ---------------------------------------------------------------------------
__global__ void dvq_reduce_kernel(const float* __restrict__ partials,
                                  float* __restrict__ out_scalar) {
    __shared__ float sm[256];
    float s = 0.f;
    for (int i = threadIdx.x; i < NPART; i += 256) s += partials[i];
    sm[threadIdx.x] = s;
    __syncthreads();
    for (int off = 128; off > 0; off >>= 1) {
        if (threadIdx.x < (unsigned)off) sm[threadIdx.x] += sm[threadIdx.x + off];
        __syncthreads();
    }
    if (threadIdx.x == 0) {
        const float scale = (1.0f + BETA) / (float)((size_t)B_ * N_ * d_);
        *out_scalar = sm[0] * scale;
    }
}

// ---------------------------------------------------------------------------
extern "C" void kernel_launch(void* const* d_in, const int* in_sizes, int n_in,
                              void* d_out, int out_size, void* d_ws, size_t ws_size,
                              hipStream_t stream) {
    const float* h  = (const float*)d_in[0];     // (8,4096,1024) f32
    const float* cb = (const float*)d_in[1];     // (4,16,256) f32

    float* out = (float*)d_out;
    float* z   = out;                                        // B*N*D floats
    int*   ids = (int*)(out + (size_t)TOKENS * D_);          // B*N int32
    float* vq  = out + (size_t)TOKENS * D_ + TOKENS;         // 1 float

    float* cnorm    = (float*)d_ws;              // 64 floats
    float* partials = cnorm + 64;                // 8192 floats

    dvq_cnorm_kernel<<<1, 64, 0, stream>>>(cb, cnorm);
    dvq_main_kernel<<<NBLOCKS, 256, 0, stream>>>(h, cb, cnorm, z, ids, partials);
    dvq_reduce_kernel<<<1, 256, 0, stream>>>(partials, vq);
}